// SpliceSitesJunctionHead_57243324121462
// MI455X (gfx1250) — compile-verified
//
#include <hip/hip_runtime.h>
#include <cstddef>

typedef __attribute__((ext_vector_type(2))) float v2f;
typedef __attribute__((ext_vector_type(4))) float v4f;
typedef __attribute__((ext_vector_type(8))) float v8f;

#define S_LEN   16384
#define D_IN    1024
#define HIDDEN  128
#define NJ      768
#define NROWS   3072   // 4 streams * 768 junctions
#define NTISS   64

// ---------------------------------------------------------------------------
// Kernel 1: gather + projection GEMM.
//   proj[g][h] = sum_k embeds[idx[g]][k] * W[k][h] + bias[h],  g in [0,3072)
// One WG = 16 gathered rows (M tile), 8 waves cover N=128 (16 each).
// K=1024 in LDS-staged chunks of 64; V_WMMA_F32_16X16X4_F32, 256 per wave.
// ---------------------------------------------------------------------------
__global__ __launch_bounds__(256) void proj_kernel(
    const float* __restrict__ embeds, const int* __restrict__ ssp,
    const float* __restrict__ W, const float* __restrict__ bias,
    float* __restrict__ proj)
{
    __shared__ float lA[16 * 68];     // 16 rows x 64 k, padded to 68 (bank-safe)
    __shared__ int   rows[16];

    const int tid  = threadIdx.x;
    const int wave = tid >> 5;
    const int lane = tid & 31;
    const int g0   = blockIdx.x * 16;

    if (tid < 16) {
        int idx = ssp[g0 + tid];                 // ssp flat: [4][768] == g
        if (idx < 0) idx = 0;
        if (idx >= S_LEN) idx = S_LEN - 1;
        rows[tid] = idx;
    }
    __syncthreads();

    const int n0     = wave * 16;
    const int m_lane = lane & 15;                // A: M index / B,C,D: N index
    const int khalf  = (lane >> 4) * 2;          // K sub-offset per half-wave

    v8f acc = {};
    for (int kc = 0; kc < D_IN; kc += 64) {
        __syncthreads();
        {   // cooperative A-chunk load: 16 rows x 64 floats, 256 B per row
            const int r = tid >> 4;              // 0..15
            const int f = (tid & 15) * 4;        // 0..60
            v4f v = *(const v4f*)(embeds + (size_t)rows[r] * D_IN + kc + f);
            *(v4f*)&lA[r * 68 + f] = v;
        }
        __syncthreads();
#pragma unroll
        for (int kk = 0; kk < 16; ++kk) {
            const int k0 = kk * 4 + khalf;
            v2f a;
            a.x = lA[m_lane * 68 + k0];
            a.y = lA[m_lane * 68 + k0 + 1];
            v2f b;
            const float* wp = W + (size_t)(kc + k0) * HIDDEN + n0 + m_lane;
            b.x = wp[0];
            b.y = wp[HIDDEN];
            acc = __builtin_amdgcn_wmma_f32_16x16x4_f32(
                false, a, false, b, (short)0, acc, false, false);
        }
    }

    const int moff = (lane >> 4) * 8;            // D: M = r + moff
    const float bv = bias[n0 + m_lane];
#pragma unroll
    for (int r = 0; r < 8; ++r)
        proj[(size_t)(g0 + moff + r) * HIDDEN + n0 + m_lane] = acc[r] + bv;
}

// ---------------------------------------------------------------------------
// Kernel 2: RoPE. One block per (stream, junction) g; 128 threads = H lanes.
//   x_t[k] = proj[g][k]*scale[t][k] + offset[t][k]
//   out[2i]   = x[2i]*cos(f_i) - x[2i+1]*sin(f_i)
//   out[2i+1] = x[2i+1]*cos(f_i) + x[2i]*sin(f_i),  f_i = idx * inv_freq[i]
// ---------------------------------------------------------------------------
__global__ __launch_bounds__(128) void rope_kernel(
    const float* __restrict__ proj, const int* __restrict__ ssp,
    const float* __restrict__ p0, const float* __restrict__ p1,
    const float* __restrict__ p2, const float* __restrict__ p3,
    float* __restrict__ rope)
{
    const int g = blockIdx.x;                    // 0..3071
    const int k = threadIdx.x;                   // 0..127
    const int s = g / NJ;
    const float* prm = (s == 0) ? p0 : (s == 1) ? p1 : (s == 2) ? p2 : p3;

    __shared__ float xp[HIDDEN];
    xp[k] = proj[(size_t)g * HIDDEN + k];

    // inv_freq[i] = 1 / (i + geomspace(1, MAX_POS-64+1, 64)[i]),
    // geomspace(1, 1048513, 64)[i] = 1048513^(i/63)
    const int   i    = k >> 1;
    const float tlin = (float)i * (1.0f / 63.0f);
    const float base = exp2f(tlin * log2f(1048513.0f));
    const float invf = 1.0f / ((float)i + base);
    const float f    = (float)ssp[g] * invf;
    float sn, cs;
    __sincosf(f, &sn, &cs);
    const float sign = (k & 1) ? 1.0f : -1.0f;
    const int   p    = k ^ 1;
    __syncthreads();

    const float xk  = xp[k];
    const float xpv = xp[p];
    float* outb = rope + (size_t)g * (NTISS * HIDDEN);
#pragma unroll 4
    for (int t = 0; t < NTISS; ++t) {
        const float xk_t = xk  * prm[t * HIDDEN + k] + prm[NTISS * HIDDEN + t * HIDDEN + k];
        const float xp_t = xpv * prm[t * HIDDEN + p] + prm[NTISS * HIDDEN + t * HIDDEN + p];
        outb[t * HIDDEN + k] = xk_t * cs + sign * xp_t * sn;  // coalesced 512B rows
    }
}

// ---------------------------------------------------------------------------
// Kernel 3: per-tissue batched GEMM + softplus + mask + channel-fastest store.
// One WG = one 16x16 (donor, acceptor) tile. 8 waves x 16 channels (c<64:
// pos tissue c, c>=64: neg tissue c-64). 32 f32 WMMAs per channel (K=128).
// Results transposed through padded LDS so global writes are 512 B bursts.
// ---------------------------------------------------------------------------
#define OUT_PAD 132                              // 128 + 4 pad (LDS banks)

__global__ __launch_bounds__(256) void junction_kernel(
    const float* __restrict__ rope, const int* __restrict__ ssp,
    float* __restrict__ out)
{
    extern __shared__ float outBuf[];            // 256 positions x OUT_PAD
    __shared__ int dmask[2][16], amask[2][16];

    const int tid  = threadIdx.x;
    const int wave = tid >> 5;
    const int lane = tid & 31;
    const int d0   = blockIdx.x * 16;
    const int a0   = blockIdx.y * 16;

    if (tid < 64) {                              // mask bits from raw indices
        const int pol  = tid >> 5;
        const int kind = (tid >> 4) & 1;
        const int r    = tid & 15;
        if (kind == 0) dmask[pol][r] = (ssp[(2 * pol)     * NJ + d0 + r] >= 0);
        else           amask[pol][r] = (ssp[(2 * pol + 1) * NJ + a0 + r] >= 0);
    }

    const int m_lane = lane & 15;
    const int khalf  = (lane >> 4) * 2;
    const int moff   = (lane >> 4) * 8;

    for (int ci = 0; ci < 16; ++ci) {
        const int c   = wave * 16 + ci;          // output channel 0..127
        const int t   = c & 63;
        const int pol = c >> 6;
        const float* Ab = rope + ((size_t)((2 * pol)     * NJ + d0 + m_lane) * NTISS + t) * HIDDEN;
        const float* Bb = rope + ((size_t)((2 * pol + 1) * NJ + a0 + m_lane) * NTISS + t) * HIDDEN;

        v8f acc = {};
#pragma unroll
        for (int kk = 0; kk < 32; ++kk) {
            const int k0 = kk * 4 + khalf;
            v2f a = *(const v2f*)(Ab + k0);
            v2f b = *(const v2f*)(Bb + k0);
            acc = __builtin_amdgcn_wmma_f32_16x16x4_f32(
                false, a, false, b, (short)0, acc, false, false);
        }
#pragma unroll
        for (int r = 0; r < 8; ++r) {            // softplus, stash transposed
            const float v  = acc[r];
            const float sp = (v > 20.0f) ? v : log1pf(__expf(v));
            outBuf[((moff + r) * 16 + m_lane) * OUT_PAD + c] = sp;
        }
    }
    __syncthreads();

    // channel-fastest coalesced store of pred and mask
    const size_t predSize = (size_t)NJ * NJ * 128;
    const int half = tid >> 7;                   // 0/1: two positions per pass
    const int c    = tid & 127;
    const int pol  = c >> 6;
#pragma unroll 4
    for (int it = 0; it < 128; ++it) {
        const int pos = it * 2 + half;           // 0..255
        const int m = pos >> 4, n = pos & 15;
        const float mk = (dmask[pol][m] && amask[pol][n]) ? 1.0f : 0.0f;
        const size_t o = ((size_t)(d0 + m) * NJ + (a0 + n)) * 128 + c;
        out[o]            = outBuf[pos * OUT_PAD + c] * mk;
        out[predSize + o] = mk;
    }
}

// ---------------------------------------------------------------------------
extern "C" void kernel_launch(void* const* d_in, const int* in_sizes, int n_in,
                              void* d_out, int out_size, void* d_ws, size_t ws_size,
                              hipStream_t stream) {
    const float* embeds = (const float*)d_in[0];   // (1,16384,1024)
    const int*   ssp    = (const int*)  d_in[1];   // (1,4,768)
    const float* W      = (const float*)d_in[2];   // (1024,128)
    const float* bias   = (const float*)d_in[3];   // (128,)
    const float* pdp    = (const float*)d_in[4];   // pos_donor  (16384,)
    const float* pap    = (const float*)d_in[5];   // pos_accept
    const float* ndp    = (const float*)d_in[6];   // neg_donor
    const float* nap    = (const float*)d_in[7];   // neg_accept
    float* out = (float*)d_out;

    // workspace: proj (3072x128 f32) then rope (3072x64x128 f32)
    float* proj = (float*)d_ws;
    float* rope = proj + (size_t)NROWS * HIDDEN;

    proj_kernel<<<NROWS / 16, 256, 0, stream>>>(embeds, ssp, W, bias, proj);
    rope_kernel<<<NROWS, 128, 0, stream>>>(proj, ssp, pdp, pap, ndp, nap, rope);

    const int dynLds = 256 * OUT_PAD * (int)sizeof(float);   // 135168 B (<320KB/WG)
    hipFuncSetAttribute(reinterpret_cast<const void*>(junction_kernel),
                        hipFuncAttributeMaxDynamicSharedMemorySize, dynLds);
    junction_kernel<<<dim3(NJ / 16, NJ / 16), 256, dynLds, stream>>>(rope, ssp, out);
}